// PhotonicMesh_54855322304738
// MI455X (gfx1250) — compile-verified
//
#include <hip/hip_runtime.h>
#include <math.h>

// Problem constants (from reference)
#define CHN      256      // N channels
#define ND       128      // double layers
#define NEV      128      // even MZIs per layer
#define NOD      127      // odd MZIs per layer
#define LSTRIDE  255      // MZIs per double layer (NE+NO)
#define BATCHSZ  16384

static constexpr float ATTEN = 0.97723722095581068f; // sqrt(10^(-0.2/10)) = 10^-0.01

typedef __attribute__((ext_vector_type(2))) float v2f;
typedef __attribute__((ext_vector_type(8))) float v8f;

#define WMMA_F32(A, B, C) \
  __builtin_amdgcn_wmma_f32_16x16x4_f32(false, (A), false, (B), (short)0, (C), false, false)

// ---------------------------------------------------------------------------
// Kernel 1: propagate the 256x256 identity through the MZI mesh to build the
// complex transfer matrix S (out_row = x_row @ S). One block per basis vector
// k; the 256-complex state lives in LDS. Writes S pair-interleaved along the
// K (reduction) dim so kernel 2's WMMA B-fragments are contiguous 8B loads:
//   SrP[(k>>1)*512 + i*2 + (k&1)] = Re S[k][i]   (same for SiP / Im)
// ---------------------------------------------------------------------------
__device__ __forceinline__ void apply_mzi(float* sre, float* sim, int i, int j,
                                          float th, float ph) {
  float s, c, epi, epr;
  sincosf(th, &s, &c);
  sincosf(ph, &epi, &epr);
  const float ar = sre[i], ai = sim[i];
  const float br = sre[j], bi = sim[j];
  const float ecr = epr * c, eci = epi * c;   // ep * cos
  const float esr = epr * s, esi = epi * s;   // ep * sin
  // na = ep*c*a - s*b ; nb = ep*s*a + c*b  (then * ATTEN)
  const float nar = ecr * ar - eci * ai - s * br;
  const float nai = ecr * ai + eci * ar - s * bi;
  const float nbr = esr * ar - esi * ai + c * br;
  const float nbi = esr * ai + esi * ar + c * bi;
  sre[i] = nar * ATTEN; sim[i] = nai * ATTEN;
  sre[j] = nbr * ATTEN; sim[j] = nbi * ATTEN;
}

__global__ void __launch_bounds__(128)
mesh_build_kernel(const float* __restrict__ thetas, const float* __restrict__ phis,
                  float* __restrict__ SrP, float* __restrict__ SiP) {
  __shared__ float sre[CHN];
  __shared__ float sim[CHN];
  const int k = blockIdx.x;    // basis vector index (K dim of the GEMM)
  const int t = threadIdx.x;   // 0..127

  sre[2 * t]     = (2 * t     == k) ? 1.0f : 0.0f;
  sre[2 * t + 1] = (2 * t + 1 == k) ? 1.0f : 0.0f;
  sim[2 * t] = 0.0f;
  sim[2 * t + 1] = 0.0f;
  __syncthreads();

  for (int d = 0; d < ND; ++d) {
    {  // even layer: pair (2t, 2t+1)
      const int pi = d * LSTRIDE + t;
      apply_mzi(sre, sim, 2 * t, 2 * t + 1, thetas[pi], phis[pi]);
    }
    __syncthreads();
    if (t < NOD) {  // odd layer: pair (2t+1, 2t+2)
      const int pi = d * LSTRIDE + NEV + t;
      apply_mzi(sre, sim, 2 * t + 1, 2 * t + 2, thetas[pi], phis[pi]);
    }
    __syncthreads();
  }

  // Emit row k of S in pair-interleaved layout.
  float* rbase = SrP + (size_t)(k >> 1) * (2 * CHN) + (k & 1);
  float* ibase = SiP + (size_t)(k >> 1) * (2 * CHN) + (k & 1);
  for (int i = t; i < CHN; i += 128) {
    rbase[2 * i] = sre[i];
    ibase[2 * i] = sim[i];
  }
}

// ---------------------------------------------------------------------------
// Kernel 2: fused GEMM + |.|^2 on the FP32 matrix cores.
//   out[b][n] = (sum_k X[b][k] Sr[k][n])^2 + (sum_k X[b][k] Si[k][n])^2
//
// Register blocking: each wave computes a 32x32 output patch = 2 M-tiles x
// 2 N-tiles x {real, imag} = 8 independent f32 accumulators, so each k-step
// issues 8 V_WMMA_F32_16X16X4_F32 from 6 b64 loads (0.75 loads/WMMA) and the
// C-operand dependency distance is 8 WMMAs. Block = 8 waves as 2x4 covering a
// 64x128 macro-tile; grid = (16384/64) x (256/128) = 512 blocks.
//
// Fragment layouts per CDNA5 ISA (32-bit operands, wave32):
//   A 16x4 : lanes 0-15 -> M=lane, {K=0,K=1}; lanes 16-31 -> M=lane-16, {K=2,K=3}
//   B 4x16 : lanes 0-15 -> N=lane, {K=0,K=1}; lanes 16-31 -> N=lane-16, {K=2,K=3}
//   C/D    : VGPR r, lanes 0-15 -> (M=r, N=lane); lanes 16-31 -> (M=r+8, N=lane-16)
// ---------------------------------------------------------------------------
__global__ void __launch_bounds__(256)
mesh_gemm_kernel(const float* __restrict__ X, const float* __restrict__ SrP,
                 const float* __restrict__ SiP, float* __restrict__ out) {
  const int lane = threadIdx.x & 31;
  const int wave = threadIdx.x >> 5;                 // 0..7
  const int wm = wave >> 2;                          // 0..1
  const int wn = wave & 3;                           // 0..3
  const int bm = blockIdx.x >> 1;                    // 0..255
  const int bn = blockIdx.x & 1;                     // 0..1
  const int row0 = bm * 64 + wm * 32;                // wave's first output row
  const int col0 = bn * 128 + wn * 32;               // wave's first output col
  const int lh = lane >> 4;                          // lane half: 0/1
  const int lm = lane & 15;

  v8f accR00 = {}, accR01 = {}, accR10 = {}, accR11 = {};
  v8f accI00 = {}, accI01 = {}, accI10 = {}, accI11 = {};

  // A: lane covers X[row + lm][k + lh*2 + {0,1}] -> contiguous float2.
  const float* aptr0 = X + (size_t)(row0 + lm) * CHN + lh * 2;        // M-tile 0
  const float* aptr1 = X + (size_t)(row0 + 16 + lm) * CHN + lh * 2;   // M-tile 1
  // B: pair q = k/2 + lh holds S[2q][col], S[2q+1][col] interleaved.
  const size_t bbase = (size_t)lh * (2 * CHN) + (col0 + lm) * 2;
  const float* brptr = SrP + bbase;                  // N-tile 0; +32 floats = N-tile 1
  const float* biptr = SiP + bbase;

#pragma unroll 2
  for (int k = 0; k < CHN; k += 4) {
    const size_t boff = (size_t)(k >> 1) * (2 * CHN);
    const v2f a0  = *(const v2f*)(aptr0 + k);
    const v2f a1  = *(const v2f*)(aptr1 + k);
    const v2f br0 = *(const v2f*)(brptr + boff);
    const v2f br1 = *(const v2f*)(brptr + boff + 32);
    const v2f bi0 = *(const v2f*)(biptr + boff);
    const v2f bi1 = *(const v2f*)(biptr + boff + 32);
    accR00 = WMMA_F32(a0, br0, accR00);
    accR01 = WMMA_F32(a0, br1, accR01);
    accR10 = WMMA_F32(a1, br0, accR10);
    accR11 = WMMA_F32(a1, br1, accR11);
    accI00 = WMMA_F32(a0, bi0, accI00);
    accI01 = WMMA_F32(a0, bi1, accI01);
    accI10 = WMMA_F32(a1, bi0, accI10);
    accI11 = WMMA_F32(a1, bi1, accI11);
  }

  // Fused squared magnitude + non-temporal store (output is never re-read;
  // keep it out of L2 so X/S stay resident).
  float* ob00 = out + (size_t)(row0 + lh * 8) * CHN + col0 + lm;
  float* ob10 = out + (size_t)(row0 + 16 + lh * 8) * CHN + col0 + lm;
#pragma unroll
  for (int r = 0; r < 8; ++r) {
    float vr, vi;
    vr = accR00[r]; vi = accI00[r];
    __builtin_nontemporal_store(vr * vr + vi * vi, ob00 + (size_t)r * CHN);
    vr = accR01[r]; vi = accI01[r];
    __builtin_nontemporal_store(vr * vr + vi * vi, ob00 + (size_t)r * CHN + 16);
    vr = accR10[r]; vi = accI10[r];
    __builtin_nontemporal_store(vr * vr + vi * vi, ob10 + (size_t)r * CHN);
    vr = accR11[r]; vi = accI11[r];
    __builtin_nontemporal_store(vr * vr + vi * vi, ob10 + (size_t)r * CHN + 16);
  }
}

// ---------------------------------------------------------------------------
extern "C" void kernel_launch(void* const* d_in, const int* in_sizes, int n_in,
                              void* d_out, int out_size, void* d_ws, size_t ws_size,
                              hipStream_t stream) {
  const float* x      = (const float*)d_in[0];  // (16384, 256) f32
  const float* thetas = (const float*)d_in[1];  // (32640,) f32
  const float* phis   = (const float*)d_in[2];  // (32640,) f32
  float* out = (float*)d_out;                   // (16384, 256) f32

  // Workspace: SrP and SiP, 256*256 floats each (512 KB total).
  float* SrP = (float*)d_ws;
  float* SiP = SrP + CHN * CHN;

  // Stage 1: build transfer matrix (256 basis vectors, one block each).
  mesh_build_kernel<<<CHN, 128, 0, stream>>>(thetas, phis, SrP, SiP);

  // Stage 2: fused WMMA GEMM + magnitude. 512 blocks x 8 waves,
  // each wave owns a 32x32 output patch.
  const int blocks = (BATCHSZ / 64) * (CHN / 128);  // 512
  mesh_gemm_kernel<<<blocks, 256, 0, stream>>>(x, SrP, SiP, out);
}